// InvariantPointAttentionMultimer_36240934044333
// MI455X (gfx1250) — compile-verified
//
#include <hip/hip_runtime.h>
#include <math.h>

// ---------------------------------------------------------------------------
// Invariant Point Attention (multimer) fused pipeline for gfx1250 (MI455X).
// Bandwidth-bound on z (302 MB fp32): fused per-q-tile kernel streams z from
// HBM once (pass-2 re-read hits the 192MB L2). All contractions use
// V_WMMA_F32_16X16X4_F32 (fp32 matrix path, wave32). LDS staging uses CDNA5
// async global->LDS (ASYNCcnt) + global_prefetch for the next tile.
// ---------------------------------------------------------------------------

#define NRES  768
#define CSD   384
#define CZD   128
#define CHD   16
#define HD    12
#define PQKD  4
#define PVD   8
#define TQ    16
#define NKT   (NRES / 16)          // 48 key tiles
#define CATD  2112                 // H*(CZ + CH + 4*PV)
#define INFV  100000.0f
#define RSQRT3 0.5773502691896258f
#define SCALW 0.25f                // sqrt(1/CH)
#define PWCONST 0.23570226039551584f // sqrt(2/(9*PQK))

typedef __attribute__((ext_vector_type(2))) float v2f;
typedef __attribute__((ext_vector_type(8))) float v8f;

static __device__ __forceinline__ v8f wmma4(v2f a, v2f b, v8f c) {
  // V_WMMA_F32_16X16X4_F32: D = A(16x4) * B(4x16) + C(16x16)
  return __builtin_amdgcn_wmma_f32_16x16x4_f32(false, a, false, b, (short)0, c,
                                               false, false);
}

// Async global->LDS 16-byte copy (GLOBAL_LOAD_ASYNC_TO_LDS_B128, ASYNCcnt).
// Generic LDS pointers carry the LDS offset in their low 32 bits (ISA 10.2).
static __device__ __forceinline__ void async_copy16(void* lds_dst,
                                                    const void* gsrc) {
  unsigned loff = (unsigned)(size_t)lds_dst;
  asm volatile("global_load_async_to_lds_b128 %0, %1, off"
               :: "v"(loff), "v"(gsrc)
               : "memory");
}
static __device__ __forceinline__ void wait_async0() {
  asm volatile("s_wait_asynccnt 0x0" ::: "memory");
}

// ---------------------------------------------------------------------------
// Generic fp32 WMMA GEMM: C[M,N] = A[M,K] @ B[K,N] (+ bias[N]).
// One wave per 16x16 output tile; M,N multiples of 16, K multiple of 4.
// ---------------------------------------------------------------------------
__global__ __launch_bounds__(256) void gemm_wmma_f32(
    const float* __restrict__ A, const float* __restrict__ B,
    const float* __restrict__ bias, float* __restrict__ C,
    int M, int N, int K) {
  const int lane = threadIdx.x & 31;
  const int wave = threadIdx.x >> 5;
  const int gw   = blockIdx.x * 8 + wave;
  const int ntiles = N >> 4;
  const int mtiles = M >> 4;
  const int mt = gw / ntiles;
  const int nt = gw % ntiles;
  if (mt >= mtiles) return;

  const int m    = lane & 15;       // A-row / B-col / D-col index for this lane
  const int hi   = lane >> 4;
  const int koff = hi * 2;

  v8f acc = {};
  const float* ap = A + (size_t)(mt * 16 + m) * K + koff;
  const float* bp = B + (size_t)koff * N + nt * 16 + m;
  const size_t bstep = (size_t)4 * N;
  for (int kb = 0; kb < K; kb += 4) {
    v2f a, b;
    a.x = ap[0];
    a.y = ap[1];
    b.x = bp[0];
    b.y = bp[N];
    acc = wmma4(a, b, acc);
    ap += 4;
    bp += bstep;
  }
  float bv = bias ? bias[nt * 16 + m] : 0.0f;
#pragma unroll
  for (int r = 0; r < 8; ++r) {
    int q = r + 8 * hi;
    C[(size_t)(mt * 16 + q) * N + nt * 16 + m] = acc[r] + bv;
  }
}

// ---------------------------------------------------------------------------
// Prep: scale q by 1/sqrt(CH); rigid-apply (R p + t) for q/k/v points;
// per-(n,h) squared norms of q/k point sets.
// raw point layout: [n][h*3*P + d*P + p]  -> pts layout [n][h][p][3]
// ---------------------------------------------------------------------------
__global__ void ipa_prep(const float* __restrict__ rot,
                         const float* __restrict__ trans,
                         float* __restrict__ qraw,      // in-place scale
                         const float* __restrict__ qpraw,
                         const float* __restrict__ kpraw,
                         const float* __restrict__ vpraw,
                         float* __restrict__ qpts, float* __restrict__ kpts,
                         float* __restrict__ vpts,
                         float* __restrict__ sqq, float* __restrict__ sqk) {
  int idx = blockIdx.x * blockDim.x + threadIdx.x;
  if (idx >= NRES * HD) return;
  int n = idx / HD, h = idx % HD;

  for (int c = 0; c < CHD; ++c)
    qraw[(size_t)n * (HD * CHD) + h * CHD + c] *= SCALW;

  float R[9], t3[3];
#pragma unroll
  for (int i = 0; i < 9; ++i) R[i] = rot[n * 9 + i];
#pragma unroll
  for (int i = 0; i < 3; ++i) t3[i] = trans[n * 3 + i];

  {
    float s = 0.f;
    for (int p = 0; p < PQKD; ++p) {
      float pl[3];
#pragma unroll
      for (int d = 0; d < 3; ++d)
        pl[d] = qpraw[(size_t)n * (HD * 3 * PQKD) + h * 3 * PQKD + d * PQKD + p];
#pragma unroll
      for (int i = 0; i < 3; ++i) {
        float g = R[i * 3] * pl[0] + R[i * 3 + 1] * pl[1] + R[i * 3 + 2] * pl[2] + t3[i];
        qpts[((size_t)n * HD + h) * (PQKD * 3) + p * 3 + i] = g;
        s += g * g;
      }
    }
    sqq[n * HD + h] = s;
  }
  {
    float s = 0.f;
    for (int p = 0; p < PQKD; ++p) {
      float pl[3];
#pragma unroll
      for (int d = 0; d < 3; ++d)
        pl[d] = kpraw[(size_t)n * (HD * 3 * PQKD) + h * 3 * PQKD + d * PQKD + p];
#pragma unroll
      for (int i = 0; i < 3; ++i) {
        float g = R[i * 3] * pl[0] + R[i * 3 + 1] * pl[1] + R[i * 3 + 2] * pl[2] + t3[i];
        kpts[((size_t)n * HD + h) * (PQKD * 3) + p * 3 + i] = g;
        s += g * g;
      }
    }
    sqk[n * HD + h] = s;
  }
  for (int p = 0; p < PVD; ++p) {
    float pl[3];
#pragma unroll
    for (int d = 0; d < 3; ++d)
      pl[d] = vpraw[(size_t)n * (HD * 3 * PVD) + h * 3 * PVD + d * PVD + p];
#pragma unroll
    for (int i = 0; i < 3; ++i) {
      float g = R[i * 3] * pl[0] + R[i * 3 + 1] * pl[1] + R[i * 3 + 2] * pl[2] + t3[i];
      vpts[((size_t)n * HD + h) * (PVD * 3) + p * 3 + i] = g;
    }
  }
}

// ---------------------------------------------------------------------------
// Fused attention kernel: one workgroup (256 thr = 8 waves) per 16-query tile.
// LDS layout (floats):
// ---------------------------------------------------------------------------
#define SM_ZT    0        // 32768 : z tile [16q][16k][128]
#define SM_QS    32768    //  3072 : q scalars [16][192]
#define SM_KV    35840    //  3072 : k (pass1) / v (pass2) scalars [16][192]
#define SM_QPT   38912    //  2304 : q_pts [16][12][12]
#define SM_KVPT  41216    //  4608 : k_pts [16][12][12] / v_pts [16][12][24]
#define SM_SQQ   45824    //   192
#define SM_SQK   46016    //   192
#define SM_LT    46208    //  3072 : bias tile [256 pairs][12]
#define SM_PT    49280    //  3072 : probs [12][16][16]
#define SM_RME   52352    //   192 : row-max encoded (uint)
#define SM_RMF   52544    //   192 : row-max float
#define SM_RSUM  52736    //   192 : row sum
#define SM_PW    52928    //    16 : per-head point weights
#define SM_MQ    52944    //    16
#define SM_MK    52960    //    16
#define SM_OPTL  52976    //  4608 : o_pt dump [16][12][24]
#define SM_TOTAL 57584    // floats -> 230336 bytes dynamic LDS

// async-stage one [16q][16k][128c] z tile; per fixed q the 16k*128c block is
// contiguous (2048 floats) in memory.
static __device__ __forceinline__ void stage_z_async(float* zt, const float* z,
                                                     int q0, int k0, int tid) {
  for (int i = tid; i < TQ * 16 * (CZD / 4); i += 256) {
    int q = i >> 9;            // 512 float4 per q row
    int off = i & 511;
    async_copy16(zt + i * 4,
                 z + ((size_t)(q0 + q) * NRES + k0) * CZD + off * 4);
  }
}

__global__ __launch_bounds__(256, 1) void ipa_attn(
    const float* __restrict__ qg, const float* __restrict__ kg,
    const float* __restrict__ vg, const float* __restrict__ qpts,
    const float* __restrict__ kpts, const float* __restrict__ vpts,
    const float* __restrict__ sqq, const float* __restrict__ sqk,
    const float* __restrict__ z, const float* __restrict__ wb,
    const float* __restrict__ bb, const float* __restrict__ mask,
    const float* __restrict__ hwts, const float* __restrict__ rot,
    const float* __restrict__ trans, float* __restrict__ Lg,
    float* __restrict__ cat) {
  extern __shared__ float sm[];
  float* zt   = sm + SM_ZT;
  float* qs   = sm + SM_QS;
  float* kv   = sm + SM_KV;
  float* qpt  = sm + SM_QPT;
  float* kvpt = sm + SM_KVPT;
  float* sqqs = sm + SM_SQQ;
  float* sqks = sm + SM_SQK;
  float* lt   = sm + SM_LT;
  float* pt   = sm + SM_PT;
  unsigned* rme = (unsigned*)(sm + SM_RME);
  float* rmf  = sm + SM_RMF;
  float* rsum = sm + SM_RSUM;
  float* pwv  = sm + SM_PW;
  float* mq   = sm + SM_MQ;
  float* mk   = sm + SM_MK;
  float* optl = sm + SM_OPTL;

  const int tid  = threadIdx.x;
  const int wave = tid >> 5;
  const int lane = tid & 31;
  const int m    = lane & 15;
  const int hi   = lane >> 4;
  const int koff = hi * 2;
  const int q0   = blockIdx.x * TQ;

  // ---- async-stage q-side tile state ----
  for (int i = tid; i < (TQ * HD * CHD) / 4; i += 256)
    async_copy16(qs + i * 4, qg + (size_t)q0 * 192 + i * 4);
  for (int i = tid; i < (TQ * HD * 12) / 4; i += 256)
    async_copy16(qpt + i * 4, qpts + (size_t)q0 * 144 + i * 4);
  if (tid < (TQ * HD) / 4)
    async_copy16(sqqs + tid * 4, sqq + q0 * HD + tid * 4);
  if (tid < TQ / 4)
    async_copy16(mq + tid * 4, mask + q0 + tid * 4);
  for (int i = tid; i < TQ * HD; i += 256) {
    rme[i] = 0u;             // below the ordered-uint code of any finite float
    rsum[i] = 0.0f;
  }
  if (tid < HD) {
    float x = hwts[tid];
    float sp = (x > 20.0f) ? x : log1pf(__expf(x));   // softplus
    pwv[tid] = PWCONST * sp;
  }
  wait_async0();
  __syncthreads();

  // =================== PASS 1: logits + row max ===================
  for (int kt = 0; kt < NKT; ++kt) {
    const int k0 = kt * 16;
    // async staging of k-side state + z tile (ASYNCcnt)
    for (int i = tid; i < (16 * HD * CHD) / 4; i += 256)
      async_copy16(kv + i * 4, kg + (size_t)k0 * 192 + i * 4);
    for (int i = tid; i < (16 * HD * 12) / 4; i += 256)
      async_copy16(kvpt + i * 4, kpts + (size_t)k0 * 144 + i * 4);
    if (tid < (16 * HD) / 4)
      async_copy16(sqks + tid * 4, sqk + k0 * HD + tid * 4);
    if (tid < 4)
      async_copy16(mk + tid * 4, mask + k0 + tid * 4);
    stage_z_async(zt, z, q0, k0, tid);

    // prefetch next z k-tile into cache while async loads are in flight
    if (kt + 1 < NKT) {
      const int k0n = k0 + 16;
#pragma unroll
      for (int j = 0; j < 4; ++j) {
        int l = tid + j * 256;           // 1024 x 128B lines
        int q = l >> 6, off = l & 63;
        __builtin_prefetch(z + ((size_t)(q0 + q) * NRES + k0n) * CZD + off * 32,
                           0, 1);
      }
    }
    wait_async0();
    __syncthreads();

    // --- phase a: pair bias  bias[pair(256), h(12)] = z_tile @ w_b + b_b ---
#pragma unroll
    for (int mi = 0; mi < 2; ++mi) {
      const int mt = wave * 2 + mi;
      v8f acc = {};
      const float* zrow = zt + (mt * 16 + m) * CZD + koff;
      const float* wrow = wb + koff * HD + m;
      const bool bn = (m < HD);
      for (int kb = 0; kb < CZD; kb += 4) {
        v2f a, b;
        a.x = zrow[0];
        a.y = zrow[1];
        b.x = bn ? wrow[0] : 0.0f;
        b.y = bn ? wrow[HD] : 0.0f;
        acc = wmma4(a, b, acc);
        zrow += 4;
        wrow += 4 * HD;
      }
      if (bn) {
        float bvb = bb[m];
#pragma unroll
        for (int r = 0; r < 8; ++r)
          lt[(mt * 16 + r + 8 * hi) * HD + m] = acc[r] + bvb;
      }
    }
    __syncthreads();

    // --- phase b: per-head logits = qk + pw*cross (+ bias, mask, consts) ---
#pragma unroll
    for (int hit = 0; hit < 2; ++hit) {
      const int hreal = wave + hit * 8;
      const bool hvalid = (hreal < HD);
      const int h = hvalid ? hreal : (HD - 1);     // clamp: redundant compute
      const float pwh = pwv[h];
      v8f acc = {};
      const float* arow = qs + m * 192 + h * 16;
      const float* brow = kv + m * 192 + h * 16;   // B is N(key)-major: same form
      for (int kb = 0; kb < 16; kb += 4) {
        v2f a, b;
        a.x = arow[kb + koff];  a.y = arow[kb + koff + 1];
        b.x = brow[kb + koff];  b.y = brow[kb + koff + 1];
        acc = wmma4(a, b, acc);
      }
      const float* ap = qpt + (m * HD + h) * 12;
      const float* bp = kvpt + (m * HD + h) * 12;
      for (int kb = 0; kb < 12; kb += 4) {         // K = 12 point dims
        v2f a, b;
        a.x = ap[kb + koff] * pwh;  a.y = ap[kb + koff + 1] * pwh;
        b.x = bp[kb + koff];        b.y = bp[kb + koff + 1];
        acc = wmma4(a, b, acc);
      }
#pragma unroll
      for (int r = 0; r < 8; ++r) {
        const int q = r + 8 * hi, kk = m;
        float l = acc[r]
                - 0.5f * pwh * (sqqs[q * HD + h] + sqks[kk * HD + h])
                + lt[(q * 16 + kk) * HD + h]
                + INFV * (mq[q] * mk[kk] - 1.0f);
        l *= RSQRT3;
        if (hvalid)
          Lg[((size_t)h * NRES + (q0 + q)) * NRES + (k0 + kk)] = l;
        unsigned u = __float_as_uint(l);
        unsigned key = (u & 0x80000000u) ? ~u : (u | 0x80000000u);
        atomicMax(&rme[q * HD + h], key);
      }
    }
    __syncthreads();
  }

  // decode row max
  for (int i = tid; i < TQ * HD; i += 256) {
    unsigned key = rme[i];
    unsigned u = (key & 0x80000000u) ? (key ^ 0x80000000u) : ~key;
    rmf[i] = __uint_as_float(u);
  }
  __syncthreads();

  // =================== PASS 2: softmax + o / o_pt / o_pair ===================
  v8f acc_o[2];  v8f acc_p[2][2];  v8f acc_z[2][8];
#pragma unroll
  for (int i = 0; i < 2; ++i) {
    acc_o[i] = (v8f){};
    acc_p[i][0] = (v8f){};  acc_p[i][1] = (v8f){};
#pragma unroll
    for (int c = 0; c < 8; ++c) acc_z[i][c] = (v8f){};
  }

  for (int kt = 0; kt < NKT; ++kt) {
    const int k0 = kt * 16;
    // async staging (z tile re-read hits L2); overlap with prob computation
    for (int i = tid; i < (16 * HD * CHD) / 4; i += 256)
      async_copy16(kv + i * 4, vg + (size_t)k0 * 192 + i * 4);
    for (int i = tid; i < (16 * HD * 24) / 4; i += 256)
      async_copy16(kvpt + i * 4, vpts + (size_t)k0 * 288 + i * 4);
    stage_z_async(zt, z, q0, k0, tid);

    // probabilities + running row sums (independent of async LDS targets)
    for (int i = tid; i < HD * 256; i += 256) {
      int h = i >> 8, rem = i & 255, q = rem >> 4, kk = rem & 15;
      float l = Lg[((size_t)h * NRES + (q0 + q)) * NRES + (k0 + kk)];
      float e = __expf(l - rmf[q * HD + h]);
      pt[i] = e;
      atomicAdd(&rsum[q * HD + h], e);
    }
    wait_async0();
    __syncthreads();

    // o and o_pt (per head)
#pragma unroll
    for (int hit = 0; hit < 2; ++hit) {
      const int hreal = wave + hit * 8;
      const int h = (hreal < HD) ? hreal : (HD - 1);
      const float* arow = pt + h * 256 + m * 16;
      for (int kb = 0; kb < 16; kb += 4) {
        v2f a, b;
        a.x = arow[kb + koff];  a.y = arow[kb + koff + 1];
        b.x = kv[(kb + koff) * 192 + h * 16 + m];
        b.y = kv[(kb + koff + 1) * 192 + h * 16 + m];
        acc_o[hit] = wmma4(a, b, acc_o[hit]);
      }
#pragma unroll
      for (int jt = 0; jt < 2; ++jt) {
        const int j0 = jt * 16 + m;
        const bool jv = (j0 < PVD * 3);
        for (int kb = 0; kb < 16; kb += 4) {
          v2f a, b;
          a.x = arow[kb + koff];  a.y = arow[kb + koff + 1];
          b.x = jv ? kvpt[((kb + koff) * HD + h) * 24 + j0] : 0.0f;
          b.y = jv ? kvpt[((kb + koff + 1) * HD + h) * 24 + j0] : 0.0f;
          acc_p[hit][jt] = wmma4(a, b, acc_p[hit][jt]);
        }
      }
    }

    // o_pair: wave owns q rows {2w, 2w+1}; A[m=h][K=key], B=z[q][key][c]
#pragma unroll
    for (int qi = 0; qi < 2; ++qi) {
      const int q = wave * 2 + qi;
      const bool av = (m < HD);
#pragma unroll
      for (int ct = 0; ct < 8; ++ct) {
        for (int kb = 0; kb < 16; kb += 4) {
          v2f a, b;
          a.x = av ? pt[m * 256 + q * 16 + kb + koff] : 0.0f;
          a.y = av ? pt[m * 256 + q * 16 + kb + koff + 1] : 0.0f;
          b.x = zt[(q * 16 + kb + koff) * CZD + ct * 16 + m];
          b.y = zt[(q * 16 + kb + koff + 1) * CZD + ct * 16 + m];
          acc_z[qi][ct] = wmma4(a, b, acc_z[qi][ct]);
        }
      }
    }
    __syncthreads();
  }

  // =================== finalize ===================
#pragma unroll
  for (int hit = 0; hit < 2; ++hit) {
    const int h = wave + hit * 8;
    if (h < HD) {
#pragma unroll
      for (int r = 0; r < 8; ++r) {
        const int q = r + 8 * hi;
        cat[(size_t)(q0 + q) * CATD + h * 16 + m] = acc_o[hit][r] / rsum[q * HD + h];
      }
#pragma unroll
      for (int jt = 0; jt < 2; ++jt) {
        const int j0 = jt * 16 + m;
        if (j0 < PVD * 3) {
#pragma unroll
          for (int r = 0; r < 8; ++r) {
            const int q = r + 8 * hi;
            optl[(q * HD + h) * 24 + j0] = acc_p[hit][jt][r] / rsum[q * HD + h];
          }
        }
      }
    }
  }
#pragma unroll
  for (int qi = 0; qi < 2; ++qi) {
    const int q = wave * 2 + qi;
#pragma unroll
    for (int ct = 0; ct < 8; ++ct) {
#pragma unroll
      for (int r = 0; r < 8; ++r) {
        const int h = r + 8 * hi;
        if (h < HD)
          cat[(size_t)(q0 + q) * CATD + 576 + h * CZD + ct * 16 + m] =
              acc_z[qi][ct][r] / rsum[q * HD + h];
      }
    }
  }
  __syncthreads();

  // o_pt -> local frame: R^T (p - t), plus norms
  for (int f = tid; f < TQ * HD * PVD; f += 256) {
    int q = f / (HD * PVD);
    int rem = f % (HD * PVD);
    int h = rem / PVD, p = rem % PVD;
    int ng = q0 + q;
    float gx = optl[(q * HD + h) * 24 + p * 3 + 0] - trans[ng * 3 + 0];
    float gy = optl[(q * HD + h) * 24 + p * 3 + 1] - trans[ng * 3 + 1];
    float gz = optl[(q * HD + h) * 24 + p * 3 + 2] - trans[ng * 3 + 2];
    float lx = rot[ng * 9 + 0] * gx + rot[ng * 9 + 3] * gy + rot[ng * 9 + 6] * gz;
    float ly = rot[ng * 9 + 1] * gx + rot[ng * 9 + 4] * gy + rot[ng * 9 + 7] * gz;
    float lz = rot[ng * 9 + 2] * gx + rot[ng * 9 + 5] * gy + rot[ng * 9 + 8] * gz;
    float n2 = lx * lx + ly * ly + lz * lz;
    float nm = sqrtf(fmaxf(n2, 1e-8f * 1e-8f));
    int mc = h * PVD + p;
    size_t base = (size_t)ng * CATD;
    cat[base + 192 + mc] = lx;
    cat[base + 288 + mc] = ly;
    cat[base + 384 + mc] = lz;
    cat[base + 480 + mc] = nm;
  }
}

// ---------------------------------------------------------------------------
// Host-side launch
// ---------------------------------------------------------------------------
extern "C" void kernel_launch(void* const* d_in, const int* in_sizes, int n_in,
                              void* d_out, int out_size, void* d_ws, size_t ws_size,
                              hipStream_t stream) {
  const float* s     = (const float*)d_in[0];
  const float* z     = (const float*)d_in[1];
  const float* mask  = (const float*)d_in[2];
  const float* rot   = (const float*)d_in[3];
  const float* trans = (const float*)d_in[4];
  const float* w_q   = (const float*)d_in[5];
  const float* w_k   = (const float*)d_in[6];
  const float* w_v   = (const float*)d_in[7];
  const float* w_qp  = (const float*)d_in[8];
  const float* b_qp  = (const float*)d_in[9];
  const float* w_kp  = (const float*)d_in[10];
  const float* b_kp  = (const float*)d_in[11];
  const float* w_vp  = (const float*)d_in[12];
  const float* b_vp  = (const float*)d_in[13];
  const float* w_b   = (const float*)d_in[14];
  const float* b_b   = (const float*)d_in[15];
  const float* hwts  = (const float*)d_in[16];
  const float* w_out = (const float*)d_in[17];
  const float* b_out = (const float*)d_in[18];
  float* out = (float*)d_out;

  // workspace layout (floats); total ~40.2 MB
  float* ws = (float*)d_ws;
  float* qraw  = ws;                 // 768*192
  float* kraw  = qraw + 147456;      // 768*192
  float* vraw  = kraw + 147456;      // 768*192
  float* qpraw = vraw + 147456;      // 768*144
  float* kpraw = qpraw + 110592;     // 768*144
  float* vpraw = kpraw + 110592;     // 768*288
  float* qpts  = vpraw + 221184;     // 768*144
  float* kpts  = qpts + 110592;      // 768*144
  float* vpts  = kpts + 110592;      // 768*288
  float* sqq   = vpts + 221184;      // 9216
  float* sqk   = sqq + 9216;         // 9216
  float* Lg    = sqk + 9216;         // 12*768*768
  float* cat   = Lg + 7077888;       // 768*2112

  // 1) projections from s (WMMA GEMMs)
  gemm_wmma_f32<<<72, 256, 0, stream>>>(s, w_q, nullptr, qraw, NRES, 192, CSD);
  gemm_wmma_f32<<<72, 256, 0, stream>>>(s, w_k, nullptr, kraw, NRES, 192, CSD);
  gemm_wmma_f32<<<72, 256, 0, stream>>>(s, w_v, nullptr, vraw, NRES, 192, CSD);
  gemm_wmma_f32<<<54, 256, 0, stream>>>(s, w_qp, b_qp, qpraw, NRES, 144, CSD);
  gemm_wmma_f32<<<54, 256, 0, stream>>>(s, w_kp, b_kp, kpraw, NRES, 144, CSD);
  gemm_wmma_f32<<<108, 256, 0, stream>>>(s, w_vp, b_vp, vpraw, NRES, 288, CSD);

  // 2) rigid apply + scaling + squared norms
  ipa_prep<<<(NRES * HD + 255) / 256, 256, 0, stream>>>(
      rot, trans, qraw, qpraw, kpraw, vpraw, qpts, kpts, vpts, sqq, sqk);

  // 3) fused attention: one workgroup per 16-query tile, z read once from HBM
  ipa_attn<<<NRES / TQ, 256, SM_TOTAL * sizeof(float), stream>>>(
      qraw, kraw, vraw, qpts, kpts, vpts, sqq, sqk, z, w_b, b_b, mask, hwts,
      rot, trans, Lg, cat);

  // 4) output projection: out = cat @ w_out + b_out
  gemm_wmma_f32<<<144, 256, 0, stream>>>(cat, w_out, b_out, out, NRES, CSD, CATD);
}